// MaskedEmbedding_64690797412603
// MI455X (gfx1250) — compile-verified
//
#include <hip/hip_runtime.h>

typedef float v2f __attribute__((ext_vector_type(2)));
typedef float v8f __attribute__((ext_vector_type(8)));
typedef unsigned int u32x4 __attribute__((ext_vector_type(4)));
typedef unsigned int u32x8 __attribute__((ext_vector_type(8)));

#define B_ 8
#define E_ 16
#define G_ 20000
#define P_ 512
#define PT 32              // p-columns per workgroup (2 waves x 16)
#define KT 80              // g-rows per chunk (20 wmma steps); 250*80 = 20000
#define DEPTH 3            // per-wave TDM ring depth (chunks in flight)
#define TILEF (KT * 16)    // floats per LDS tile (1280)
#define TILEB (TILEF * 4)  // bytes per LDS tile (5120)
#define NCHUNK (G_ / KT)   // 250

// Per-wave 2D TDM tile load: KT rows x 16 floats from a row-major [.,512]
// f32 tensor into LDS at byte offset lds_off (tile stored densely, row
// stride 16 floats -> conflict-free fragment reads, no padding needed).
template <bool NT>
__device__ __forceinline__ void tdm_issue(unsigned lds_off, const float* gsrc) {
    unsigned long long ga = (unsigned long long)(uintptr_t)gsrc;
    u32x4 g0;
    g0[0] = 1u;                               // count=1, user descriptor
    g0[1] = lds_off;                          // LDS byte address
    g0[2] = (unsigned)ga;                     // global_addr[31:0]
    g0[3] = ((unsigned)(ga >> 32) & 0x1FFFFFFu) | (2u << 30); // [56:32], type=2
    u32x8 g1;
    g1[0] = (2u << 16);                       // data_size = 4 bytes, no padding
    g1[1] = (unsigned)P_ << 16;               // tensor_dim0 = 512 (bits 63:48)
    g1[2] = (unsigned)G_ << 16;               // tensor_dim1 = 20000 (bits 95:80)
    g1[3] = 16u << 16;                        // tile_dim0 = 16 (bits 127:112)
    g1[4] = (unsigned)KT;                     // tile_dim1 = 80
    g1[5] = (unsigned)P_;                     // tensor_dim0_stride = 512
    g1[6] = 0u;
    g1[7] = 0u;
    if (NT)
        asm volatile("tensor_load_to_lds %0, %1 th:TH_LOAD_NT"
                     :: "s"(g0), "s"(g1) : "memory");
    else
        asm volatile("tensor_load_to_lds %0, %1"
                     :: "s"(g0), "s"(g1) : "memory");
}

// out[b,e,p] = sum_g x[b,g] * weight[e,g,p] * mask[g,p] + bias[e,p]
// Per-e GEMM C[8,512] = x[8,G] @ (W_e*mask)[G,512] on v_wmma_f32_16x16x4_f32.
// Each wave runs a private, barrier-free TDM->LDS->WMMA pipeline over its
// own 16-column slice, synchronized only by its own TENSORcnt.
__global__ __launch_bounds__(64) void masked_embed_wmma_tdm(
    const float* __restrict__ x,     // [B,G]
    const float* __restrict__ w,     // [E,G,P]
    const float* __restrict__ bias,  // [E,P]
    const float* __restrict__ mask,  // [G,P]
    float* __restrict__ out)         // [B,E,P]
{
    extern __shared__ float lds[];   // 2 waves * DEPTH * 2 tiles, offset 0

    const int e       = blockIdx.x;          // 0..15
    const int pbase   = blockIdx.y * PT;     // 0,32,..,480
    const int wave    = threadIdx.x >> 5;    // 0..1
    const int lane    = threadIdx.x & 31;
    const int nloc    = lane & 15;
    const int khalf   = (lane >> 4) << 1;    // 0 or 2
    const int colbase = pbase + wave * 16;   // wave's 16-column slice
    const int col     = colbase + nloc;      // global p column

    const float* __restrict__ xR = x + (size_t)(nloc & 7) * G_;
    const float* __restrict__ wE = w + (size_t)e * G_ * P_;

    const unsigned ldsW = (unsigned)(wave * (DEPTH * 2 * TILEB)); // byte base
    const float*   myL  = lds + (size_t)wave * (DEPTH * 2 * TILEF);

    v8f c = {0.f, 0.f, 0.f, 0.f, 0.f, 0.f, 0.f, 0.f};

    // Prologue: fill DEPTH-1 ring slots.
#pragma unroll
    for (int cc = 0; cc < DEPTH - 1; ++cc) {
        const size_t gOff = (size_t)cc * KT * P_ + colbase;
        tdm_issue<true >(ldsW + (2 * cc + 0) * TILEB, &wE[gOff]);
        tdm_issue<false>(ldsW + (2 * cc + 1) * TILEB, &mask[gOff]);
    }

    for (int ci = 0; ci < NCHUNK; ++ci) {
        const int g = ci * KT;

        // Refill the slot this wave finished reading last iteration.
        const int cn = ci + DEPTH - 1;
        asm volatile("s_wait_dscnt 0x0" ::: "memory"); // ring slot reads drained
        if (cn < NCHUNK) {
            const int bn = cn % DEPTH;
            const size_t gOff = (size_t)cn * KT * P_ + colbase;
            tdm_issue<true >(ldsW + (2 * bn + 0) * TILEB, &wE[gOff]);
            tdm_issue<false>(ldsW + (2 * bn + 1) * TILEB, &mask[gOff]);
        }
        // TDM completes in order per wave: <= 2*(DEPTH-1) outstanding
        // implies chunk ci's two tiles are in LDS. No cross-wave sync.
        __builtin_amdgcn_s_wait_tensorcnt(2 * (DEPTH - 1));

        const int buf = ci % DEPTH;
        const float* Wt = myL + (2 * buf + 0) * TILEF;
        const float* Mt = myL + (2 * buf + 1) * TILEF;

#pragma unroll
        for (int kk = 0; kk < KT; kk += 4) {
            const int r = kk + khalf;        // K row for fragment element .x

            v2f a;                           // A: x[b, g+r], x[b, g+r+1]
            a.x = xR[g + r];
            a.y = xR[g + r + 1];

            v2f b;                           // B: masked weight from LDS
            b.x = Wt[r * 16 + nloc]       * Mt[r * 16 + nloc];
            b.y = Wt[(r + 1) * 16 + nloc] * Mt[(r + 1) * 16 + nloc];

            c = __builtin_amdgcn_wmma_f32_16x16x4_f32(
                    /*neg_a=*/false, a, /*neg_b=*/false, b,
                    /*c_mod=*/(short)0, c, /*reuse_a=*/false, /*reuse_b=*/false);
        }
    }

    // C layout: VGPR r -> M=r (lanes 0-15) / M=r+8 (lanes 16-31); only b=0..7.
    if (lane < 16) {
        const float bv = bias[e * P_ + col];
#pragma unroll
        for (int r = 0; r < 8; ++r) {
            out[((size_t)r * E_ + e) * P_ + col] = c[r] + bv;
        }
    }
}

extern "C" void kernel_launch(void* const* d_in, const int* in_sizes, int n_in,
                              void* d_out, int out_size, void* d_ws, size_t ws_size,
                              hipStream_t stream) {
    const float* x    = (const float*)d_in[0];
    const float* wgt  = (const float*)d_in[1];
    const float* bias = (const float*)d_in[2];
    const float* mask = (const float*)d_in[3];
    float* out        = (float*)d_out;

    dim3 grid(E_, P_ / PT);                        // 16 x 16 workgroups
    dim3 block(64);                                // 2 waves
    const size_t shmem = (size_t)2 * DEPTH * 2 * TILEB;  // 61440 bytes
    masked_embed_wmma_tdm<<<grid, block, shmem, stream>>>(x, wgt, bias, mask, out);
}